// Net_14070312862437
// MI455X (gfx1250) — compile-verified
//
#include <hip/hip_runtime.h>
#include <hip/hip_bf16.h>

// ---------------------------------------------------------------------------
// CDNA5 (gfx1250) implementation of the field-conv point network.
// Heavy math (NN-search distance matrices, conv einsums) runs on
// V_WMMA_F32_16X16X4_F32 (wave32, one wave per block so EXEC is all-1s).
// ---------------------------------------------------------------------------

typedef float v2f __attribute__((ext_vector_type(2)));
typedef float v8f __attribute__((ext_vector_type(8)));

__device__ __forceinline__ v8f wmma_f32_k4(v2f a, v2f b, v8f c) {
    // D = A(16x4) * B(4x16) + C, fp32 throughout.
    return __builtin_amdgcn_wmma_f32_16x16x4_f32(
        /*neg_a=*/false, a, /*neg_b=*/false, b,
        /*c_mod=*/(short)0, c, /*reuse_a=*/false, /*reuse_b=*/false);
}

// ---------------------------------------------------------------------------
// Kernel 1: nearest-neighbor argmin via WMMA.
// One wave handles 16 sample points (fixed offset k) against all N points.
// A operand holds -2*sp so the WMMA emits -2*sp.xyz directly; dist =
// |sp|^2 + |x|^2 + wmma. Per-lane running min, then shfl_xor tree reduce.
// A layout: lanes 0-15 -> (x,y) of row lo, lanes 16-31 -> (z, 0) of row lo.
// B layout: lanes 0-15 -> (x,y) of col lo, lanes 16-31 -> (z, 0) of col lo.
// D layout: VGPR r -> row r + 8*(lane>>4), col = lane&15.
// ---------------------------------------------------------------------------
__global__ __launch_bounds__(32)
void fc_argmin_wmma(const float* __restrict__ x, int N, int sIn,
                    const float* __restrict__ offs, int K, int M, float edge,
                    int* __restrict__ idx)
{
    const int lane = threadIdx.x;
    const int lo = lane & 15;
    const int hi = lane >> 4;
    const int m0 = blockIdx.x * 16;
    const int k  = blockIdx.y;
    const int b  = blockIdx.z;

    const float ox = edge * offs[3 * k + 0];
    const float oy = edge * offs[3 * k + 1];
    const float oz = edge * offs[3 * k + 2];
    const float* xb = x + (size_t)b * N * sIn;

    int mr = m0 + lo; if (mr >= M) mr = M - 1;   // clamp; invalid rows never stored
    const float sx = xb[(size_t)mr * sIn + 0] + ox;
    const float sy = xb[(size_t)mr * sIn + 1] + oy;
    const float sz = xb[(size_t)mr * sIn + 2] + oz;

    __shared__ float s_spsq[16];
    if (lane < 16) s_spsq[lane] = sx * sx + sy * sy + sz * sz;
    __syncthreads();

    float spsq[8];
#pragma unroll
    for (int r = 0; r < 8; ++r) spsq[r] = s_spsq[r + 8 * hi];

    v2f a;
    a.x = hi ? (-2.0f * sz) : (-2.0f * sx);
    a.y = hi ? 0.0f         : (-2.0f * sy);

    float minv[8]; int mini[8];
#pragma unroll
    for (int r = 0; r < 8; ++r) { minv[r] = 3.0e38f; mini[r] = 0; }

#pragma unroll 2
    for (int n0 = 0; n0 < N; n0 += 16) {
        const int n = n0 + lo;
        float px = 0.f, py = 0.f, pz = 0.f, xsq = 1.0e30f;  // pad cols lose argmin
        if (n < N) {
            px = xb[(size_t)n * sIn + 0];
            py = xb[(size_t)n * sIn + 1];
            pz = xb[(size_t)n * sIn + 2];
            xsq = px * px + py * py + pz * pz;
        }
        if (n0 + 16 < N) {  // uniform guard: prefetch next tile (global_prefetch_b8)
            __builtin_prefetch(&xb[(size_t)(n0 + 16 + lo) * sIn], 0, 3);
        }
        v2f bv;
        bv.x = hi ? pz  : px;
        bv.y = hi ? 0.f : py;
        v8f c = {};
        c = wmma_f32_k4(a, bv, c);
#pragma unroll
        for (int r = 0; r < 8; ++r) {
            float d = c[r] + spsq[r] + xsq;
            if (d < minv[r]) { minv[r] = d; mini[r] = n; }
        }
    }

    // Reduce the 16 columns held by lanes {0..15} (rows 0-7) and {16..31}
    // (rows 8-15); xor offsets < 16 never cross the halves.
#pragma unroll
    for (int off = 8; off >= 1; off >>= 1) {
#pragma unroll
        for (int r = 0; r < 8; ++r) {
            float ov = __shfl_xor(minv[r], off);
            int   oi = __shfl_xor(mini[r], off);
            if (ov < minv[r] || (ov == minv[r] && oi < mini[r])) {
                minv[r] = ov; mini[r] = oi;
            }
        }
    }
    if (lo == 0) {
#pragma unroll
        for (int r = 0; r < 8; ++r) {
            int m = m0 + r + 8 * hi;
            if (m < M) idx[((size_t)b * M + m) * K + k] = mini[r];
        }
    }
}

// ---------------------------------------------------------------------------
// Kernel 2: gather + einsum 'bmkc,kcd->bmd' (+bias, ReLU) via WMMA.
// GEMM: (16 rows m) x (K*Cin) against W[(k*Cin+c), d].
// Main path: 16 kc per iteration = 4 WMMAs. Since Cin is a multiple of 16 for
// every layer but L0, a 16-kc block lives inside one k -> one idx load and one
// gather-row base serve all 4 WMMAs. lc = log2(Cin) (Cin is a power of two),
// so k = kc >> lc, c = kc & (Cin-1). Remainder path (only L0, K*Cin = 9)
// zero-pads to a multiple of 4.
// ---------------------------------------------------------------------------
__global__ __launch_bounds__(32)
void fc_conv_wmma(const float* __restrict__ x, int N, int sIn,
                  const int* __restrict__ idx, int K, int lc,
                  const float* __restrict__ W, const float* __restrict__ bias,
                  int Cout, float* __restrict__ xout, int M)
{
    const int lane = threadIdx.x;
    const int lo = lane & 15;
    const int hi = lane >> 4;
    const int m0 = blockIdx.x * 16;
    const int d0 = blockIdx.y * 16;
    const int b  = blockIdx.z;
    const int sOut = 3 + Cout;

    const float* xb  = x + (size_t)b * N * sIn;
    const int* idxb  = idx + (size_t)b * M * K;
    const int KC = K << lc;            // K * Cin
    const int cmask = (1 << lc) - 1;   // Cin - 1

    int mA = m0 + lo; if (mA >= M) mA = M - 1;
    const int d = d0 + lo;

    v8f acc = {};
    int kc0 = 0;
    for (; kc0 + 16 <= KC; kc0 += 16) {
        const int k = kc0 >> lc;
        const int j = idxb[(size_t)mA * K + k];
        const float* grow = xb + (size_t)j * sIn + 3 + (kc0 & cmask);
        const float* wrow = W + (size_t)kc0 * Cout + d;
#pragma unroll
        for (int t = 0; t < 4; ++t) {
            const int cc = 4 * t + 2 * hi;  // this lane supplies kc rows cc, cc+1
            v2f a, bv;
            a.x  = grow[cc];
            a.y  = grow[cc + 1];
            bv.x = wrow[(size_t)cc * Cout];
            bv.y = wrow[(size_t)(cc + 1) * Cout];
            acc = wmma_f32_k4(a, bv, acc);
        }
    }
    for (; kc0 < KC; kc0 += 4) {        // remainder: only L0 (KC = 9)
        const int kcp = kc0 + 2 * hi;
        float a0 = 0.f, a1 = 0.f, b0 = 0.f, b1 = 0.f;
        if (kcp < KC) {
            int j = idxb[(size_t)mA * K + (kcp >> lc)];
            a0 = xb[(size_t)j * sIn + 3 + (kcp & cmask)];
            b0 = W[(size_t)kcp * Cout + d];
        }
        if (kcp + 1 < KC) {
            int j = idxb[(size_t)mA * K + ((kcp + 1) >> lc)];
            a1 = xb[(size_t)j * sIn + 3 + ((kcp + 1) & cmask)];
            b1 = W[(size_t)(kcp + 1) * Cout + d];
        }
        v2f a, bv;
        a.x = a0; a.y = a1;
        bv.x = b0; bv.y = b1;
        acc = wmma_f32_k4(a, bv, acc);
    }

    float* xob = xout + (size_t)b * M * sOut;
    const float bia = bias[d];
#pragma unroll
    for (int r = 0; r < 8; ++r) {
        int m = m0 + r + 8 * hi;
        if (m < M) {
            float v = acc[r] + bia;
            xob[(size_t)m * sOut + 3 + d] = v > 0.f ? v : 0.f;
        }
    }
}

// ---------------------------------------------------------------------------
// Kernel 3: copy xyz centers (first 3 channels of first M rows) to new buffer.
// ---------------------------------------------------------------------------
__global__ void fc_copy_centers(const float* __restrict__ x, int N, int sIn,
                                float* __restrict__ xout, int M, int sOut, int B)
{
    int t = blockIdx.x * blockDim.x + threadIdx.x;
    int total = B * M * 3;
    if (t >= total) return;
    int b = t / (M * 3);
    int rem = t - b * (M * 3);
    int m = rem / 3, c = rem - m * 3;
    xout[((size_t)b * M + m) * sOut + c] = x[((size_t)b * N + m) * sIn + c];
}

// ---------------------------------------------------------------------------
// Kernel 4: field pool: 4 nearest points per center, channelwise max.
// ---------------------------------------------------------------------------
__global__ void fc_pool(const float* __restrict__ x, int N, int sIn, int C,
                        float* __restrict__ xout, int M)
{
    int m = blockIdx.x * blockDim.x + threadIdx.x;
    int b = blockIdx.y;
    if (m >= M) return;
    const float* xb = x + (size_t)b * N * sIn;
    float cx = xb[(size_t)m * sIn + 0];
    float cy = xb[(size_t)m * sIn + 1];
    float cz = xb[(size_t)m * sIn + 2];
    float bd[4] = {1e30f, 1e30f, 1e30f, 1e30f};
    int   bi[4] = {0, 0, 0, 0};
    for (int n = 0; n < N; ++n) {
        float dx = xb[(size_t)n * sIn + 0] - cx;
        float dy = xb[(size_t)n * sIn + 1] - cy;
        float dz = xb[(size_t)n * sIn + 2] - cz;
        float dd = dx * dx + dy * dy + dz * dz;
        if (dd < bd[3]) {
            int p = 3;
            while (p > 0 && dd < bd[p - 1]) { bd[p] = bd[p - 1]; bi[p] = bi[p - 1]; --p; }
            bd[p] = dd; bi[p] = n;
        }
    }
    const int sOut = 3 + C;
    float* xo = xout + ((size_t)b * M + m) * sOut;
    xo[0] = cx; xo[1] = cy; xo[2] = cz;
    for (int c = 0; c < C; ++c) {
        float v = xb[(size_t)bi[0] * sIn + 3 + c];
        for (int q = 1; q < 4; ++q) {
            float w = xb[(size_t)bi[q] * sIn + 3 + c];
            v = w > v ? w : v;
        }
        xo[3 + c] = v;
    }
}

// ---------------------------------------------------------------------------
// Kernel 5: dense FC (+ optional eval-mode BN + ReLU). B=2 rows -> VALU.
// ---------------------------------------------------------------------------
__global__ void fc_dense(const float* __restrict__ h, int Fin,
                         const float* __restrict__ W, const float* __restrict__ bias,
                         const float* __restrict__ g, const float* __restrict__ beta,
                         int relu, float* __restrict__ out, int Fout)
{
    int j = blockIdx.x * blockDim.x + threadIdx.x;
    int b = blockIdx.y;
    if (j >= Fout) return;
    const float* hb = h + (size_t)b * Fin;
    float acc = bias[j];
    for (int i = 0; i < Fin; ++i) acc = fmaf(hb[i], W[(size_t)i * Fout + j], acc);
    if (g) {
        const float s = 0.9999950000374997f;  // 1/sqrt(1 + 1e-5)
        acc = acc * (g[j] * s) + beta[j];
    }
    if (relu) acc = acc > 0.f ? acc : 0.f;
    out[(size_t)b * Fout + j] = acc;
}

// ---------------------------------------------------------------------------
// Orchestration. Input order (setup_inputs dict insertion order):
//   0: xyz_sdf (2,3136,4)
//   1..14:  conv_offsets[0..13]   15..28: conv_W[0..13]   29..42: conv_b[0..13]
//   43: fc1_w 44: fc1_b 45: bn1_g 46: bn1_b
//   47: fc2_w 48: fc2_b 49: bn2_g 50: bn2_b
//   51: fc3_w 52: fc3_b
// ---------------------------------------------------------------------------
extern "C" void kernel_launch(void* const* d_in, const int* in_sizes, int n_in,
                              void* d_out, int out_size, void* d_ws, size_t ws_size,
                              hipStream_t stream)
{
    static const int   KS_[14]   = {9,9,9,9,9,9,9,9,9,9,9,9,9,1};
    static const int   LC_[14]   = {0,4,4,5,5,6,6,6,7,7,7,7,7,7};  // log2(Cin)
    static const int   COUT_[14] = {16,16,32,32,64,64,64,128,128,128,128,128,128,256};
    static const int   MS_[14]   = {3136,3136,784,784,196,196,196,49,49,49,12,12,12,1};
    static const int   POOLM_[14]= {0,784,0,196,0,0,49,0,0,12,0,0,3,0};
    static const float EDGE_[14] = {
        0.10714285714285715f, 0.10714285714285715f,
        0.21428571428571430f, 0.21428571428571430f,
        0.42857142857142860f, 0.42857142857142860f, 0.42857142857142860f,
        0.85714285714285720f, 0.85714285714285720f, 0.85714285714285720f,
        1.71428571428571430f, 1.71428571428571430f, 1.71428571428571430f,
        2.0f };
    const int B = 2;

    // Workspace partition (~2.4 MB total)
    float* bufA = (float*)d_ws;                 // 262144 floats (max need 119168)
    float* bufB = bufA + (1 << 18);             // 262144 floats
    int*   idxb = (int*)(bufB + (1 << 18));     // 65536 ints (max need 56448)
    float* h1   = (float*)(idxb + (1 << 16));   // 1024 floats
    float* h2   = h1 + 1024;                    // 512 floats

    const float* cur = (const float*)d_in[0];
    int N = 3136, Cin = 1;
    float* other = bufA;

    for (int i = 0; i < 14; ++i) {
        const int K = KS_[i], M = MS_[i], Cout = COUT_[i];
        const int sIn = 3 + Cin, sOut = 3 + Cout;
        const float* offs = (const float*)d_in[1 + i];
        const float* W    = (const float*)d_in[15 + i];
        const float* bias = (const float*)d_in[29 + i];

        dim3 gA((M + 15) / 16, K, B);
        fc_argmin_wmma<<<gA, 32, 0, stream>>>(cur, N, sIn, offs, K, M, EDGE_[i], idxb);

        float* nxt = other;
        int tot = B * M * 3;
        fc_copy_centers<<<(tot + 127) / 128, 128, 0, stream>>>(cur, N, sIn, nxt, M, sOut, B);

        dim3 gC((M + 15) / 16, Cout / 16, B);
        fc_conv_wmma<<<gC, 32, 0, stream>>>(cur, N, sIn, idxb, K, LC_[i], W, bias, Cout, nxt, M);

        other = (nxt == bufA) ? bufB : bufA;
        cur = nxt; N = M; Cin = Cout;

        if (POOLM_[i]) {
            int Mp = POOLM_[i];
            float* pout = other;
            dim3 gP((Mp + 63) / 64, B);
            fc_pool<<<gP, 64, 0, stream>>>(cur, N, 3 + Cin, Cin, pout, Mp);
            other = (pout == bufA) ? bufB : bufA;
            cur = pout; N = Mp;
        }
    }

    // MLP head: (B,259) -> 512 -> 256 -> 40
    const float* fc1w = (const float*)d_in[43];
    const float* fc1b = (const float*)d_in[44];
    const float* bn1g = (const float*)d_in[45];
    const float* bn1b = (const float*)d_in[46];
    const float* fc2w = (const float*)d_in[47];
    const float* fc2b = (const float*)d_in[48];
    const float* bn2g = (const float*)d_in[49];
    const float* bn2b = (const float*)d_in[50];
    const float* fc3w = (const float*)d_in[51];
    const float* fc3b = (const float*)d_in[52];

    fc_dense<<<dim3((512 + 63) / 64, B), 64, 0, stream>>>(cur, 259, fc1w, fc1b, bn1g, bn1b, 1, h1, 512);
    fc_dense<<<dim3((256 + 63) / 64, B), 64, 0, stream>>>(h1, 512, fc2w, fc2b, bn2g, bn2b, 1, h2, 256);
    fc_dense<<<dim3((40 + 63) / 64, B), 64, 0, stream>>>(h2, 256, fc3w, fc3b, nullptr, nullptr, 0, (float*)d_out, 40);
}